// GINSmall_36764920054229
// MI455X (gfx1250) — compile-verified
//
#include <hip/hip_runtime.h>
#include <hip/hip_bf16.h>

typedef float v2f __attribute__((ext_vector_type(2)));
typedef float v8f __attribute__((ext_vector_type(8)));

// ---------------------------------------------------------------- utilities

__global__ __launch_bounds__(256) void k_copy4(float4* __restrict__ d,
                                               const float4* __restrict__ s,
                                               int n4) {
  int i = blockIdx.x * blockDim.x + threadIdx.x;
  if (i < n4) d[i] = s[i];
}

__global__ __launch_bounds__(256) void k_zero(float* __restrict__ p, int n) {
  int i = blockIdx.x * blockDim.x + threadIdx.x;
  if (i < n) p[i] = 0.0f;
}

// ------------------------------------------------- edge scatter-add (GIN agg)
// z[dst] += h[src], z pre-initialized to h so z = h + sum_{j->i} h_j.
template <int D>
__global__ __launch_bounds__(256) void k_edge_agg(float* __restrict__ z,
                                                  const float* __restrict__ h,
                                                  const int* __restrict__ src,
                                                  const int* __restrict__ dst,
                                                  int E) {
  constexpr int CH = D / 4;  // float4 chunks per row
  int idx = blockIdx.x * blockDim.x + threadIdx.x;
  if (idx >= E * CH) return;
  int e = idx / CH;
  int c = (idx - e * CH) * 4;
  int s = src[e];
  int d = dst[e];
  const float4 v = *(const float4*)(h + (size_t)s * D + c);
  float* zp = z + (size_t)d * D + c;
  atomicAdd(zp + 0, v.x);
  atomicAdd(zp + 1, v.y);
  atomicAdd(zp + 2, v.z);
  atomicAdd(zp + 3, v.w);
}

// ------------------------------------------------------------- WMMA GIN MLP
// Hout = relu( relu(Z[16xD] @ W1[DxH] + b1) @ W2[HxH] + b2 )   (H == 64)
// One wave32 per 16-row tile; 8 waves (256 threads) per workgroup.
template <int D>
__global__ __launch_bounds__(256) void k_mlp(const float* __restrict__ Z,
                                             const float* __restrict__ W1,
                                             const float* __restrict__ B1,
                                             const float* __restrict__ W2,
                                             const float* __restrict__ B2,
                                             float* __restrict__ Hout,
                                             int ntiles) {
  constexpr int H = 64;
  constexpr int MLD = H + 1;  // LDS row stride padded -> conflict-free column reads
  __shared__ float mid[8][16 * MLD];

  const int wave = threadIdx.x >> 5;
  const int lane = threadIdx.x & 31;
  const int tile = blockIdx.x * 8 + wave;
  if (tile >= ntiles) return;  // wave-uniform: EXEC stays all-ones per wave

  const int row0 = tile * 16;
  const int half = lane >> 4;  // 0: lanes 0-15 (K+0,K+1), 1: lanes 16-31 (K+2,K+3)
  const int m    = lane & 15;  // A-row / B-col / D-col index

  const float* zrow = Z + (size_t)(row0 + m) * D;

  // ---- GEMM1: mid = relu(Z @ W1 + b1) ----
  for (int nt = 0; nt < H / 16; ++nt) {
    v8f acc = {};
#pragma unroll
    for (int k = 0; k < D; k += 4) {
      v2f a, b;
      a.x = zrow[k + 2 * half + 0];
      a.y = zrow[k + 2 * half + 1];
      const float* w = W1 + (size_t)(k + 2 * half) * H + nt * 16 + m;
      b.x = w[0];
      b.y = w[H];
      acc = __builtin_amdgcn_wmma_f32_16x16x4_f32(false, a, false, b,
                                                  (short)0, acc, false, false);
    }
    const float bias = B1[nt * 16 + m];
#pragma unroll
    for (int r = 0; r < 8; ++r) {
      float v = acc[r] + bias;
      v = v > 0.0f ? v : 0.0f;
      mid[wave][(r + 8 * half) * MLD + nt * 16 + m] = v;  // row-major 16xH tile
    }
  }
  // Same-wave LDS RAW: DS ops are in-order per wave; compiler inserts s_wait_dscnt.

  // ---- GEMM2: Hout = relu(mid @ W2 + b2) ----
  for (int nt = 0; nt < H / 16; ++nt) {
    v8f acc = {};
#pragma unroll
    for (int k = 0; k < H; k += 4) {
      v2f a, b;
      a.x = mid[wave][m * MLD + k + 2 * half + 0];
      a.y = mid[wave][m * MLD + k + 2 * half + 1];
      const float* w = W2 + (size_t)(k + 2 * half) * H + nt * 16 + m;
      b.x = w[0];
      b.y = w[H];
      acc = __builtin_amdgcn_wmma_f32_16x16x4_f32(false, a, false, b,
                                                  (short)0, acc, false, false);
    }
    const float bias = B2[nt * 16 + m];
#pragma unroll
    for (int r = 0; r < 8; ++r) {
      float v = acc[r] + bias;
      v = v > 0.0f ? v : 0.0f;
      Hout[(size_t)(row0 + r + 8 * half) * H + nt * 16 + m] = v;
    }
  }
}

// --------------------------------------------------------------- mean pool
__global__ __launch_bounds__(256) void k_pool(const float* __restrict__ h,
                                              const int* __restrict__ batch,
                                              float* __restrict__ pool,
                                              float* __restrict__ cnt,
                                              int N, int H) {
  int idx = blockIdx.x * blockDim.x + threadIdx.x;
  if (idx >= N * H) return;
  int i = idx / H;
  int d = idx - i * H;
  int g = batch[i];
  atomicAdd(&pool[g * H + d], h[idx]);
  if (d == 0) atomicAdd(&cnt[g], 1.0f);
}

// ------------------------------------------------------------------ readout
// d_out = [ out (G*C) | g (G*H) ]
__global__ __launch_bounds__(256) void k_readout(const float* __restrict__ pool,
                                                 const float* __restrict__ cnt,
                                                 const float* __restrict__ Wl,
                                                 const float* __restrict__ bl,
                                                 float* __restrict__ out,
                                                 float* __restrict__ gout,
                                                 int G, int H, int C) {
  int tid = blockIdx.x * blockDim.x + threadIdx.x;
  int stride = gridDim.x * blockDim.x;
  for (int i = tid; i < G * H; i += stride) {
    int gi = i / H;
    float cc = cnt[gi];
    cc = cc > 1.0f ? cc : 1.0f;
    gout[i] = pool[i] / cc;
  }
  if (tid < G * C) {
    int gi = tid / C;
    int c = tid - gi * C;
    float cc = cnt[gi];
    cc = cc > 1.0f ? cc : 1.0f;
    float s = bl[c];
    for (int d = 0; d < H; ++d) s += (pool[gi * H + d] / cc) * Wl[d * C + c];
    out[tid] = s;
  }
}

// -------------------------------------------------------------------- launch
extern "C" void kernel_launch(void* const* d_in, const int* in_sizes, int n_in,
                              void* d_out, int out_size, void* d_ws, size_t ws_size,
                              hipStream_t stream) {
  const float* x    = (const float*)d_in[0];
  const int*   eidx = (const int*)d_in[1];
  const int*   batch= (const int*)d_in[2];
  const float* W1_0 = (const float*)d_in[3];
  const float* b1_0 = (const float*)d_in[4];
  const float* W2_0 = (const float*)d_in[5];
  const float* b2_0 = (const float*)d_in[6];
  const float* W1_1 = (const float*)d_in[7];
  const float* b1_1 = (const float*)d_in[8];
  const float* W2_1 = (const float*)d_in[9];
  const float* b2_1 = (const float*)d_in[10];
  const float* W1_2 = (const float*)d_in[11];
  const float* b1_2 = (const float*)d_in[12];
  const float* W2_2 = (const float*)d_in[13];
  const float* b2_2 = (const float*)d_in[14];
  const float* Wl   = (const float*)d_in[15];
  const float* bl   = (const float*)d_in[16];

  const int H  = in_sizes[4];           // 64
  const int IN = in_sizes[3] / H;       // 128
  const int N  = in_sizes[0] / IN;      // 50000
  const int E  = in_sizes[1] / 2;       // 800000
  const int C  = in_sizes[16];          // 2
  const int G  = out_size / (C + H);    // 64

  const int* src = eidx;
  const int* dst = eidx + E;

  // workspace layout (floats)
  float* ws   = (float*)d_ws;
  float* z    = ws;                        // N*IN (reused as N*H for layers 1-2)
  float* hA   = z + (size_t)N * IN;        // N*H
  float* hB   = hA + (size_t)N * H;        // N*H
  float* pool = hB + (size_t)N * H;        // G*H
  float* cntb = pool + (size_t)G * H;      // G

  const int TB = 256;
  const int ntiles = N / 16;               // 50000/16 = 3125 (exact)
  const int mlp_blocks = (ntiles + 7) / 8;

  // ---------------- layer 0 (D = IN) ----------------
  {
    int n4 = (N * IN) / 4;
    k_copy4<<<(n4 + TB - 1) / TB, TB, 0, stream>>>((float4*)z, (const float4*)x, n4);
    int nthr = E * (IN / 4);
    k_edge_agg<128><<<(nthr + TB - 1) / TB, TB, 0, stream>>>(z, x, src, dst, E);
    k_mlp<128><<<mlp_blocks, TB, 0, stream>>>(z, W1_0, b1_0, W2_0, b2_0, hA, ntiles);
  }
  // ---------------- layer 1 (D = H) ----------------
  {
    int n4 = (N * H) / 4;
    k_copy4<<<(n4 + TB - 1) / TB, TB, 0, stream>>>((float4*)z, (const float4*)hA, n4);
    int nthr = E * (H / 4);
    k_edge_agg<64><<<(nthr + TB - 1) / TB, TB, 0, stream>>>(z, hA, src, dst, E);
    k_mlp<64><<<mlp_blocks, TB, 0, stream>>>(z, W1_1, b1_1, W2_1, b2_1, hB, ntiles);
  }
  // ---------------- layer 2 (D = H) ----------------
  {
    int n4 = (N * H) / 4;
    k_copy4<<<(n4 + TB - 1) / TB, TB, 0, stream>>>((float4*)z, (const float4*)hB, n4);
    int nthr = E * (H / 4);
    k_edge_agg<64><<<(nthr + TB - 1) / TB, TB, 0, stream>>>(z, hB, src, dst, E);
    k_mlp<64><<<mlp_blocks, TB, 0, stream>>>(z, W1_2, b1_2, W2_2, b2_2, hA, ntiles);
  }
  // ---------------- mean pool + readout ----------------
  {
    int nz = G * H + G;
    k_zero<<<(nz + TB - 1) / TB, TB, 0, stream>>>(pool, nz);  // pool then cntb (contiguous)
    int nthr = N * H;
    k_pool<<<(nthr + TB - 1) / TB, TB, 0, stream>>>(hA, batch, pool, cntb, N, H);
    float* out_logits = (float*)d_out;           // G*C
    float* out_g      = out_logits + G * C;      // G*H
    k_readout<<<16, TB, 0, stream>>>(pool, cntb, Wl, bl, out_logits, out_g, G, H, C);
  }
}